// MKLSAGEInference_26087631356381
// MI455X (gfx1250) — compile-verified
//
#include <hip/hip_runtime.h>

// ---------------------------------------------------------------------------
// GraphSAGE inference for MI455X (gfx1250, wave32, WMMA).
//   out = (scatter_mean over edges of (x @ W_l^T + b_l)) + x @ W_r^T
// Pipeline:
//   1) zero agg/deg scratch
//   2) GEMM1: x_l = x @ W_l^T + b_l          (v_wmma_f32_16x16x4_f32)
//   3) scatter: agg[dst] += x_l[src], deg[dst] += 1   (global_atomic_add_f32)
//   4) GEMM2: out = agg/max(deg,1) + x @ W_r^T (WMMA, C preloaded with mean)
//
// LDS holds W in a k-pair-interleaved transposed layout so each lane's
// B fragment {W[n][k], W[n][k+1]} is one aligned ds_load_b64 (no VGPR
// shuffle movs in front of the WMMAs).
// ---------------------------------------------------------------------------

typedef __attribute__((ext_vector_type(2))) float v2f;
typedef __attribute__((ext_vector_type(8))) float v8f;

#define FEAT 128
#define TILE_N 8          // 8 n-tiles of 16 cols = 128 output features per wave
#define WAVES_PER_BLOCK 8 // block covers 8*16 = 128 rows

__global__ void sage_zero_kernel(float* __restrict__ p, size_t n) {
  size_t i = (size_t)blockIdx.x * blockDim.x + threadIdx.x;
  const size_t stride = (size_t)gridDim.x * blockDim.x;
  for (; i < n; i += stride) p[i] = 0.0f;
}

// One wave per edge: 32 lanes x float4 = 128 floats.
__global__ __launch_bounds__(256) void sage_scatter_kernel(
    const long long* __restrict__ ei,   // [2, n_edges] int64
    const float* __restrict__ xl,       // [n_nodes, 128]
    float* __restrict__ agg,            // [n_nodes, 128]
    float* __restrict__ deg,            // [n_nodes]
    int n_edges)
{
  const int edge = (int)(((size_t)blockIdx.x * blockDim.x + threadIdx.x) >> 5);
  const int lane = threadIdx.x & 31;
  if (edge >= n_edges) return;
  const long long src = ei[edge];                      // uniform per wave
  const long long dst = ei[(size_t)n_edges + edge];    // uniform per wave
  const float4 v = *(const float4*)(xl + (size_t)src * FEAT + lane * 4);
  float* ap = agg + (size_t)dst * FEAT + lane * 4;
  atomicAdd(ap + 0, v.x);
  atomicAdd(ap + 1, v.y);
  atomicAdd(ap + 2, v.z);
  atomicAdd(ap + 3, v.w);
  if (lane == 0) atomicAdd(deg + dst, 1.0f);
}

// Y = X @ W^T (+ bias broadcast)  OR  Y = aggr/max(deg,1) + X @ W^T
// Per wave: 16(M) x 128(N) tile, K looped in steps of 4 with fp32 WMMA.
__global__ __launch_bounds__(256) void sage_gemm_kernel(
    const float* __restrict__ X,     // [nrows, 128]
    const float* __restrict__ W,     // [128, 128] torch-style [out, in]
    const float* __restrict__ bias,  // [128] or nullptr
    const float* __restrict__ aggr,  // [nrows, 128] or nullptr
    const float* __restrict__ deg,   // [nrows] (valid when aggr != nullptr)
    float* __restrict__ Y,           // [nrows, 128]
    int nrows)
{
  // k-pair interleaved W^T: plane p = k>>1 (256 floats each),
  //   sW[p*256 + n*2 + (k&1)] = W[n][k].   64 planes * 256 floats = 64 KB.
  __shared__ __align__(16) float sW[64 * 256];
  const int tid = threadIdx.x;
  for (int i = tid; i < FEAT * (FEAT / 4); i += 256) {
    const int row = i & 127;        // W out-row (n)
    const int c4  = (i >> 7) << 2;  // W in-col group (k), multiple of 4
    const float4 v = *(const float4*)(W + row * FEAT + c4);
    // two contiguous 8B pair-writes; consecutive tid -> consecutive row*2
    // -> lanes cover all 64 banks pairwise (conflict-free)
    float2* p0 = (float2*)(sW + ((c4 >> 1) + 0) * 256 + row * 2);
    float2* p1 = (float2*)(sW + ((c4 >> 1) + 1) * 256 + row * 2);
    *p0 = make_float2(v.x, v.y);
    *p1 = make_float2(v.z, v.w);
  }
  __syncthreads();

  const int lane = tid & 31;
  const int wave = tid >> 5;
  const int l16  = lane & 15;
  const int half = lane >> 4;   // 0: K+0/+1 rows M..M+7 ; 1: K+2/+3 rows M+8..

  const int m0 = (blockIdx.x * WAVES_PER_BLOCK + wave) * 16;

  // A fragment source row for this lane (clamped; OOB rows never stored)
  const int arow  = m0 + l16;
  const int arowc = arow < nrows ? arow : 0;
  const float* aptr = X + (size_t)arowc * FEAT + half * 2;

  // B fragment base: plane (k0>>1)+half, offset (n*16+l16)*2 — one b64 per frag
  const float* bbase = sW + half * 256 + l16 * 2;

  v8f acc[TILE_N];
  if (aggr) {
    // C := agg / max(deg, 1)   per the C/D layout: VGPR i -> row m0+i+half*8
    int rowc[8];
    float scale[8];
#pragma unroll
    for (int i = 0; i < 8; ++i) {
      const int r = m0 + i + half * 8;
      rowc[i] = r < nrows ? r : 0;
      scale[i] = 1.0f / fmaxf(deg[rowc[i]], 1.0f);
    }
#pragma unroll
    for (int n = 0; n < TILE_N; ++n) {
#pragma unroll
      for (int i = 0; i < 8; ++i)
        acc[n][i] = aggr[(size_t)rowc[i] * FEAT + n * 16 + l16] * scale[i];
    }
  } else {
#pragma unroll
    for (int n = 0; n < TILE_N; ++n) {
      const float b = bias ? bias[n * 16 + l16] : 0.0f;
#pragma unroll
      for (int i = 0; i < 8; ++i) acc[n][i] = b;
    }
  }

  for (int k0 = 0; k0 < FEAT; k0 += 4) {
    v2f a;
    a.x = aptr[k0];       // K = k0 + half*2
    a.y = aptr[k0 + 1];   // K = k0 + half*2 + 1
    const float* bk = bbase + (k0 >> 1) * 256;
#pragma unroll
    for (int n = 0; n < TILE_N; ++n) {
      const v2f b = *(const v2f*)(bk + n * 32);   // single ds_load_b64
      acc[n] = __builtin_amdgcn_wmma_f32_16x16x4_f32(
          /*neg_a=*/false, a, /*neg_b=*/false, b,
          /*c_mod=*/(short)0, acc[n], /*reuse_a=*/false, /*reuse_b=*/false);
    }
  }

#pragma unroll
  for (int n = 0; n < TILE_N; ++n) {
#pragma unroll
    for (int i = 0; i < 8; ++i) {
      const int row = m0 + i + half * 8;
      if (row < nrows) Y[(size_t)row * FEAT + n * 16 + l16] = acc[n][i];
    }
  }
}

extern "C" void kernel_launch(void* const* d_in, const int* in_sizes, int n_in,
                              void* d_out, int out_size, void* d_ws, size_t ws_size,
                              hipStream_t stream) {
  const float*     x   = (const float*)d_in[0];      // [N, 128] fp32
  const long long* ei  = (const long long*)d_in[1];  // [2, E] int64
  const float*     W_l = (const float*)d_in[2];      // [128, 128]
  const float*     b_l = (const float*)d_in[3];      // [128]
  const float*     W_r = (const float*)d_in[4];      // [128, 128]
  float*           out = (float*)d_out;              // [N, 128]

  const int n_nodes = in_sizes[0] / FEAT;  // 100000
  const int n_edges = in_sizes[1] / 2;     // 1600000

  // Workspace layout: x_l | agg | deg
  float* xl  = (float*)d_ws;
  float* agg = xl + (size_t)n_nodes * FEAT;
  float* deg = agg + (size_t)n_nodes * FEAT;

  // 1) zero agg + deg (contiguous)
  const size_t nz = (size_t)n_nodes * FEAT + (size_t)n_nodes;
  sage_zero_kernel<<<2048, 256, 0, stream>>>(agg, nz);

  // 2) x_l = x @ W_l^T + b_l
  const int mblocks = (n_nodes + WAVES_PER_BLOCK * 16 - 1) / (WAVES_PER_BLOCK * 16);
  sage_gemm_kernel<<<mblocks, 256, 0, stream>>>(x, W_l, b_l, nullptr, nullptr, xl, n_nodes);

  // 3) scatter-add over edges (one wave per edge)
  const int sblocks = (n_edges + WAVES_PER_BLOCK - 1) / WAVES_PER_BLOCK;
  sage_scatter_kernel<<<sblocks, 256, 0, stream>>>(ei, xl, agg, deg, n_edges);

  // 4) out = agg/max(deg,1) + x @ W_r^T
  sage_gemm_kernel<<<mblocks, 256, 0, stream>>>(x, W_r, nullptr, agg, deg, out, n_nodes);
}